// TripletUncertaintyLoss_73607149519054
// MI455X (gfx1250) — compile-verified
//
#include <hip/hip_runtime.h>
#include <math.h>

// ---------------------------------------------------------------------------
// TripletUncertaintyLoss — masked log-sum reduction over a 4096x8192 f32 array
//
// Roofline: 134 MB single-pass read / 23.3 TB/s HBM ~= 5.8 us. Bandwidth
// bound; VALU (v_log_f32) cost is negligible. So: wide NT global loads,
// per-thread f32 accumulation, wave32 shuffle reduce, per-block partials,
// then a single-wave WMMA-based final reduce (CDNA5 v_wmma_f32_16x16x4_f32).
// ---------------------------------------------------------------------------

typedef __attribute__((ext_vector_type(4))) float vf4;
typedef __attribute__((ext_vector_type(2))) float v2f;
typedef __attribute__((ext_vector_type(8))) float v8f;

#define LOG_2PI_PLUS_1 2.8378770664093453  // ln(2*pi) + 1

// ---------------------------------------------------------------------------
// Kernel 1: streaming partial log-sum. One f32 partial per block into d_ws.
// ---------------------------------------------------------------------------
__global__ __launch_bounds__(256) void logsum_partial_kernel(
    const float* __restrict__ x, float* __restrict__ partial,
    long long n4, long long n) {
  const vf4* __restrict__ x4 = (const vf4*)x;

  long long gid    = (long long)blockIdx.x * blockDim.x + threadIdx.x;
  long long stride = (long long)gridDim.x * blockDim.x;

  float acc = 0.0f;
  for (long long i = gid; i < n4; i += stride) {
    // Single-pass streaming data: non-temporal B128 load (th:NT), keep L2
    // clean — 134 MB would otherwise churn the 192 MB L2 for nothing.
    vf4 v = __builtin_nontemporal_load(&x4[i]);
    acc += (v.x != 0.0f) ? __logf(v.x) : 0.0f;
    acc += (v.y != 0.0f) ? __logf(v.y) : 0.0f;
    acc += (v.z != 0.0f) ? __logf(v.z) : 0.0f;
    acc += (v.w != 0.0f) ? __logf(v.w) : 0.0f;
  }

  // Tail (n not multiple of 4): handled by one thread; n=33.5M is divisible
  // by 4 so this loop is empty in practice.
  if (gid == 0) {
    for (long long i = n4 * 4; i < n; ++i) {
      float t = x[i];
      acc += (t != 0.0f) ? __logf(t) : 0.0f;
    }
  }

  // wave32 butterfly reduction (CDNA5 is wave32-only)
#pragma unroll
  for (int off = 16; off > 0; off >>= 1)
    acc += __shfl_xor(acc, off, 32);

  __shared__ float wsum[8];  // 256 threads = 8 waves
  const int wave = threadIdx.x >> 5;
  if ((threadIdx.x & 31) == 0) wsum[wave] = acc;
  __syncthreads();

  if (threadIdx.x == 0) {
    float s = 0.0f;
    const int nw = blockDim.x >> 5;
    for (int w = 0; w < nw; ++w) s += wsum[w];
    partial[blockIdx.x] = s;  // every slot [0, gridDim.x) is written
  }
}

// ---------------------------------------------------------------------------
// Kernel 2: one wave (32 lanes). Lane-strided accumulate of block partials,
// then cross-lane reduce via V_WMMA_F32_16X16X4_F32 with an all-ones B:
//   A[M][0] = p[M] (lanes 0-15, a[0]),  A[M][2] = p[M+16] (lanes 16-31, a[0])
//   => D[M][N] = p[M] + p[M+16]  for every column N (exact, layout-robust
//      since B is all ones). Lane L<16 holds D[0..7][L] in d[0..7]; lane
//      L>=16 holds D[8..15][L-16]. Sum of 8 D regs + shfl_xor(16) = total.
// ---------------------------------------------------------------------------
__global__ void logsum_final_kernel(const float* __restrict__ partial,
                                    int nparts, float* __restrict__ out,
                                    long long n_elems) {
  const int lane = threadIdx.x;  // launched with exactly 32 threads, EXEC all-1s

  float acc = 0.0f;
  for (int i = lane; i < nparts; i += 32) acc += partial[i];

  v2f a; a[0] = acc;  a[1] = 0.0f;   // 16x4 f32 A fragment
  v2f b; b[0] = 1.0f; b[1] = 1.0f;   // 4x16 all-ones B fragment
  v8f c = {};
  // (neg_a, A, neg_b, B, c_mod, C, reuse_a, reuse_b)
  v8f d = __builtin_amdgcn_wmma_f32_16x16x4_f32(
      false, a, false, b, (short)0, c, false, false);

  float s = d[0] + d[1] + d[2] + d[3] + d[4] + d[5] + d[6] + d[7];
  s += __shfl_xor(s, 16, 32);  // combine M=0..7 half with M=8..15 half

  if (lane == 0) {
    // total = (m*n/2) * (ln(2pi)+1) + 0.5 * sumD ; out = total>0 ? log1p : 1
    double total = (double)n_elems * 0.5 * LOG_2PI_PLUS_1 + 0.5 * (double)s;
    out[0] = (total > 0.0) ? (float)log1p(total) : 1.0f;
  }
}

// ---------------------------------------------------------------------------
// Launch
// ---------------------------------------------------------------------------
extern "C" void kernel_launch(void* const* d_in, const int* in_sizes, int n_in,
                              void* d_out, int out_size, void* d_ws, size_t ws_size,
                              hipStream_t stream) {
  const float* x = (const float*)d_in[0];
  const long long n = (long long)in_sizes[0];
  const long long n4 = n / 4;

  float* partial = (float*)d_ws;

  // 2048 blocks x 256 threads -> 524288 threads, 16 float4 loads each:
  // plenty of in-flight B128 loads to saturate 23.3 TB/s. Clamp to the
  // workspace capacity (one f32 partial per block).
  int nBlocks = 2048;
  long long maxB = (long long)(ws_size / sizeof(float));
  if ((long long)nBlocks > maxB) nBlocks = (int)maxB;
  if (nBlocks < 1) nBlocks = 1;

  logsum_partial_kernel<<<nBlocks, 256, 0, stream>>>(x, partial, n4, n);
  logsum_final_kernel<<<1, 32, 0, stream>>>(partial, nBlocks, (float*)d_out, n);
}